// SimpleGaussianModel_7902739825355
// MI455X (gfx1250) — compile-verified
//
#include <hip/hip_runtime.h>
#include <hip/hip_bf16.h>
#include <math.h>

// ---------------------------------------------------------------------------
// Gaussian point-splat rasterizer for MI455X (gfx1250, wave32).
//  - projection via V_WMMA_F32_16X16X4_F32 (f32 needed: fp16 ulp @1024 = 1.0)
//  - z-buffer via global_atomic_min_u32 (positive floats order as uints)
//  - bilinear scatter via hardware f32 atomic adds (stays in 192MB L2)
// ---------------------------------------------------------------------------

static constexpr int H = 1080;
static constexpr int W = 1920;
static constexpr int NPIX = H * W;

typedef float v2f __attribute__((ext_vector_type(2)));
typedef float v8f __attribute__((ext_vector_type(8)));

// ---------------------------------------------------------------------------
// Per-wave projection through WMMA.
// A (16x4 f32, 2 VGPRs): lanes 0-15 carry K=0 (x) / K=1 (y); lanes 16-31
// carry K=2 (z) / K=3 (1.0). Two issues cover the wave's 32 gaussians.
// B (4x16 f32, 2 VGPRs): VGPR0 = rows K=0 (lanes 0-15) / K=2 (lanes 16-31),
// VGPR1 = rows K=1 / K=3;  B[k][n] = M[n][k], M = viewmat rows 0..2 (row 3 = 0).
// D (16x16 f32, 8 VGPRs): element (m,n) -> VGPR m%8, lane n + 16*(m/8).
// Only columns n<3 are live; they are exchanged through LDS back to
// one-gaussian-per-lane form. LDS ops of one wave are in-order, and we add a
// block barrier for safety (no divergence on this path; EXEC all-ones at WMMA).
// ---------------------------------------------------------------------------
__device__ __forceinline__ void project_wave(float mx, float my, float mz,
                                             const float* Mlds,      // [16]: Mlds[k*4+n] = M[n][k]
                                             float (*ex)[4],         // this wave's [32][4] exchange
                                             float& mc0, float& mc1, float& mc2) {
  const unsigned l  = threadIdx.x & 31u;
  const unsigned lo = l & 15u;
  const bool     hi = l >= 16u;

  v2f B;
  B.x = (lo < 4u) ? Mlds[(hi ? 2 : 0) * 4 + lo] : 0.0f;   // row K = hi?2:0
  B.y = (lo < 4u) ? Mlds[(hi ? 3 : 1) * 4 + lo] : 0.0f;   // row K = hi?3:1

#pragma unroll
  for (int c = 0; c < 2; ++c) {
    const int src = (int)lo + (c ? 16 : 0);
    const float sx = __shfl(mx, src, 32);
    const float sy = __shfl(my, src, 32);
    const float sz = __shfl(mz, src, 32);
    v2f A;
    A.x = hi ? sz : sx;      // K=0 (x) for lanes 0-15, K=2 (z) for lanes 16-31
    A.y = hi ? 1.0f : sy;    // K=1 (y)                 K=3 (1)
    v8f acc = {};
    acc = __builtin_amdgcn_wmma_f32_16x16x4_f32(false, A, false, B,
                                                (short)0, acc, false, false);
    if (lo < 3u) {           // lanes holding live columns n = 0,1,2
      const int mbase = (hi ? 8 : 0) + (c ? 16 : 0);
#pragma unroll
      for (int r = 0; r < 8; ++r) ex[mbase + r][lo] = acc[r];
    }
  }
  __syncthreads();
  mc0 = ex[l][0];
  mc1 = ex[l][1];
  mc2 = ex[l][2];
}

// Shared per-kernel preamble: build Mlds[k*4+n] = viewmat[n*4+k] (n<3), else 0.
#define BUILD_MLDS()                                        \
  __shared__ float Mlds[16];                                \
  __shared__ float ex[8][32][4];                            \
  {                                                         \
    const int t = threadIdx.x;                              \
    if (t < 16) {                                           \
      const int k = t >> 2, nn = t & 3;                     \
      Mlds[t] = (nn < 3) ? viewmat[nn * 4 + k] : 0.0f;      \
    }                                                       \
    __syncthreads();                                        \
  }

__global__ void __launch_bounds__(256) init_kernel(float* __restrict__ buf,
                                                   unsigned* __restrict__ zbuf) {
  const int stride = gridDim.x * blockDim.x;
  int i = blockIdx.x * blockDim.x + threadIdx.x;
  for (int p = i; p < NPIX * 5; p += stride) buf[p] = 0.0f;
  for (int p = i; p < NPIX; p += stride) zbuf[p] = 0x7F800000u;  // +inf
}

__global__ void __launch_bounds__(256) zmin_kernel(const float* __restrict__ means,
                                                   const float* __restrict__ viewmat,
                                                   const float* __restrict__ Kmat,
                                                   unsigned* __restrict__ zbuf, int n) {
  BUILD_MLDS();
  const int g = blockIdx.x * 256 + threadIdx.x;
  const bool valid = g < n;
  float mx = 0.f, my = 0.f, mz = 0.f;
  if (valid) { mx = means[3 * g]; my = means[3 * g + 1]; mz = means[3 * g + 2]; }

  float mc0, mc1, mc2;
  project_wave(mx, my, mz, Mlds, ex[threadIdx.x >> 5], mc0, mc1, mc2);

  const float fx = Kmat[0], cx = Kmat[2], fy = Kmat[4], cy = Kmat[5];
  const float z = mc2;
  const bool front = z > 0.1f;
  const float zs = front ? z : 1.0f;
  const float x = mc0 * fx / zs + cx;
  const float y = mc1 * fy / zs + cy;
  const bool on = valid && front && (x >= 0.f) && (x < (float)(W - 1)) &&
                  (y >= 0.f) && (y < (float)(H - 1));
  if (on) {
    int x0 = (int)floorf(x);
    int y0 = (int)floorf(y);
    x0 = min(max(x0, 0), W - 1);
    y0 = min(max(y0, 0), H - 1);
    atomicMin(&zbuf[y0 * W + x0], __float_as_uint(z));  // z>0.1 => uint-orderable
  }
}

__device__ __forceinline__ void corner_add(float* __restrict__ b,
                                           float r, float g, float bl, float z, float w) {
  unsafeAtomicAdd(b + 0, r * w);
  unsafeAtomicAdd(b + 1, g * w);
  unsafeAtomicAdd(b + 2, bl * w);
  unsafeAtomicAdd(b + 3, z * w);
  unsafeAtomicAdd(b + 4, w);
}

__global__ void __launch_bounds__(256) scatter_kernel(const float* __restrict__ means,
                                                      const float* __restrict__ colors,
                                                      const float* __restrict__ viewmat,
                                                      const float* __restrict__ Kmat,
                                                      const unsigned* __restrict__ zbuf,
                                                      float* __restrict__ buf, int n) {
  BUILD_MLDS();
  const int g = blockIdx.x * 256 + threadIdx.x;
  const bool valid = g < n;
  float mx = 0.f, my = 0.f, mz = 0.f;
  if (valid) { mx = means[3 * g]; my = means[3 * g + 1]; mz = means[3 * g + 2]; }

  float mc0, mc1, mc2;
  project_wave(mx, my, mz, Mlds, ex[threadIdx.x >> 5], mc0, mc1, mc2);

  const float fx = Kmat[0], cx = Kmat[2], fy = Kmat[4], cy = Kmat[5];
  const float z = mc2;
  const bool front = z > 0.1f;
  const float zs = front ? z : 1.0f;
  const float x = mc0 * fx / zs + cx;
  const float y = mc1 * fy / zs + cy;
  const bool on = valid && front && (x >= 0.f) && (x < (float)(W - 1)) &&
                  (y >= 0.f) && (y < (float)(H - 1));
  if (!on) return;

  const float x0f = floorf(x), y0f = floorf(y);
  const float dx = x - x0f, dy = y - y0f;
  int x0 = min(max((int)x0f, 0), W - 1);
  int y0 = min(max((int)y0f, 0), H - 1);
  const int x1 = min(x0 + 1, W - 1);
  const int y1 = min(y0 + 1, H - 1);
  const int pix = y0 * W + x0;

  const float zb = __uint_as_float(zbuf[pix]);
  if (!(z <= zb + 0.05f)) return;  // vis

  const float cr = 1.0f / (1.0f + expf(-colors[3 * g]));
  const float cg = 1.0f / (1.0f + expf(-colors[3 * g + 1]));
  const float cb = 1.0f / (1.0f + expf(-colors[3 * g + 2]));

  const float wa = (1.f - dx) * (1.f - dy);
  const float wb = dx * (1.f - dy);
  const float wc = (1.f - dx) * dy;
  const float wd = dx * dy;

  // NOTE: corner pairing matches the reference exactly (wb->(x0,y1), wc->(x1,y0)).
  corner_add(buf + (size_t)pix * 5,            cr, cg, cb, z, wa);
  corner_add(buf + (size_t)(y1 * W + x0) * 5,  cr, cg, cb, z, wb);
  corner_add(buf + (size_t)(y0 * W + x1) * 5,  cr, cg, cb, z, wc);
  corner_add(buf + (size_t)(y1 * W + x1) * 5,  cr, cg, cb, z, wd);
}

__global__ void __launch_bounds__(256) norm_kernel(const float* __restrict__ buf,
                                                   float* __restrict__ out) {
  const int p = blockIdx.x * 256 + threadIdx.x;
  if (p >= NPIX) return;
  const float* b = buf + (size_t)p * 5;
  const float tot = b[4] + 1e-6f;
  const float r = fminf(fmaxf(b[0] / tot, 0.f), 1.f);
  const float g = fminf(fmaxf(b[1] / tot, 0.f), 1.f);
  const float bl = fminf(fmaxf(b[2] / tot, 0.f), 1.f);
  const float d = b[3] / tot;
  float* o = out + (size_t)p * 4;
  o[0] = r; o[1] = g; o[2] = bl; o[3] = d;
}

extern "C" void kernel_launch(void* const* d_in, const int* in_sizes, int n_in,
                              void* d_out, int out_size, void* d_ws, size_t ws_size,
                              hipStream_t stream) {
  const float* means   = (const float*)d_in[0];
  const float* colors  = (const float*)d_in[1];
  // d_in[2] opacities, d_in[3] scales, d_in[4] quats: unused by reference output
  const float* viewmat = (const float*)d_in[5];
  const float* Kmat    = (const float*)d_in[6];
  const int n = in_sizes[0] / 3;

  float*    buf  = (float*)d_ws;      // NPIX*5 f32 accumulation (41.5 MB, fits L2)
  unsigned* zbuf = (unsigned*)d_out;  // stage z-buffer in d_out's first NPIX words

  init_kernel<<<2048, 256, 0, stream>>>(buf, zbuf);
  const int gb = (n + 255) / 256;
  zmin_kernel<<<gb, 256, 0, stream>>>(means, viewmat, Kmat, zbuf, n);
  scatter_kernel<<<gb, 256, 0, stream>>>(means, colors, viewmat, Kmat,
                                         (const unsigned*)zbuf, buf, n);
  norm_kernel<<<(NPIX + 255) / 256, 256, 0, stream>>>(buf, (float*)d_out);
}